// DynamicCondLinear_78357383348735
// MI455X (gfx1250) — compile-verified
//
#include <hip/hip_runtime.h>

// Problem constants (from reference): B=32, CB=1, IN=2048, OUT=2048, K=8, H=512
#define B_   32
#define IN_  2048
#define OUT_ 2048
#define K_   8
#define H_   512

typedef __attribute__((ext_vector_type(2))) float v2f;
typedef __attribute__((ext_vector_type(8))) float v8f;

// Workspace layout (floats):
//   [0..7]            : a[8]        (softmax mixture weights)
//   [8..2055]         : agg_b[2048] (Σ_k a_k * kernels_bias[k][o])
//   [2056..2056+8192) : h partials  (16 blocks x 512)
#define WS_A     0
#define WS_AGGB  8
#define WS_HPART 2056

// ---------------------------------------------------------------------------
// Stage 1: partial h[j] = Σ_{i in slice} cond[i] * w1[i][j]   (16 i-slices)
// ---------------------------------------------------------------------------
__global__ __launch_bounds__(512) void stage1_hpart(const float* __restrict__ cond,
                                                    const float* __restrict__ w1,
                                                    float* __restrict__ ws) {
  const int j    = threadIdx.x;          // 0..511
  const int g    = blockIdx.x;           // 0..15
  const int ibeg = g * (IN_ / 16);
  float acc = 0.f;
  #pragma unroll 4
  for (int i = ibeg; i < ibeg + IN_ / 16; ++i)
    acc = fmaf(cond[i], w1[(size_t)i * H_ + j], acc);
  ws[WS_HPART + g * H_ + j] = acc;
}

// ---------------------------------------------------------------------------
// Stage 2: h = relu(Σ partials + b1); scores = h@w2 + b2; a = softmax(scores);
//          agg_b[o] = Σ_k a_k * kernels_bias[k][o]
// ---------------------------------------------------------------------------
__global__ __launch_bounds__(512) void stage2_scores(const float* __restrict__ b1,
                                                     const float* __restrict__ w2,
                                                     const float* __restrict__ b2,
                                                     const float* __restrict__ kb,
                                                     float* __restrict__ ws) {
  __shared__ float h[H_];
  __shared__ float sc[K_];
  __shared__ float al[K_];
  const int j = threadIdx.x;

  float acc = b1[j];
  #pragma unroll
  for (int g = 0; g < 16; ++g) acc += ws[WS_HPART + g * H_ + j];
  h[j] = fmaxf(acc, 0.f);
  __syncthreads();

  if (j < K_) {
    float s = b2[j];
    for (int t = 0; t < H_; ++t) s = fmaf(h[t], w2[t * K_ + j], s);
    sc[j] = s;
  }
  __syncthreads();

  if (j == 0) {
    float m = sc[0];
    #pragma unroll
    for (int k = 1; k < K_; ++k) m = fmaxf(m, sc[k]);
    float e[K_], sum = 0.f;
    #pragma unroll
    for (int k = 0; k < K_; ++k) { e[k] = __expf(sc[k] - m); sum += e[k]; }
    const float inv = 1.f / sum;
    #pragma unroll
    for (int k = 0; k < K_; ++k) { al[k] = e[k] * inv; ws[WS_A + k] = al[k]; }
  }
  __syncthreads();

  for (int o = j; o < OUT_; o += H_) {
    float bb = 0.f;
    #pragma unroll
    for (int k = 0; k < K_; ++k) bb = fmaf(al[k], kb[k * OUT_ + o], bb);
    ws[WS_AGGB + o] = bb;
  }
}

// ---------------------------------------------------------------------------
// Stage 3: outT[o,b] = Σ_i (Σ_k a_k W_k[o,i]) * x[b,i]  via f32 WMMA 16x16x4.
// Per WG: 16-row o-tile (grid = 128 WGs), K=2048 in 256-wide i-blocks staged
// through LDS with float4 (global_load_b128 / ds_store_b128). 8 waves slice
// the 256 staged columns 8-ways; deterministic LDS reduction at the end.
// ---------------------------------------------------------------------------
#define OTILE    16
#define IBLK     256
#define LDSTRIDE (IBLK + 4)   // stride 260 floats: 16B-aligned rows for b128,
                              // bank stride 4 (mod 64) -> conflict-free b64 reads

__global__ __launch_bounds__(256) void moe_gemm(const float* __restrict__ x,
                                                const float* __restrict__ kw,
                                                const float* __restrict__ ws,
                                                float* __restrict__ out) {
  __shared__ float aW[OTILE][LDSTRIDE];   // aggregated weight tile (o x i)
  __shared__ float xs[B_][LDSTRIDE];      // x tile (b x i)
  __shared__ float red[8 * 512];          // per-wave partial D tiles
  __shared__ float a_sm[K_];

  const int tid   = threadIdx.x;
  const int oBase = blockIdx.x * OTILE;
  if (tid < K_) a_sm[tid] = ws[WS_A + tid];
  __syncthreads();

  const int wv   = tid >> 5;              // wave 0..7
  const int ln   = tid & 31;
  const int row  = ln & 15;               // A/B: row within lane group
  const int sel  = ln >> 4;               // K-pair select (ISA 32-bit A layout)
  const int ks   = wv;                    // K slice: columns ks*32..+31
  // staging geometry: 64 lanes per row-slice, 4 floats per lane
  const int srow = tid >> 6;              // 0..3
  const int scol = (tid & 63) * 4;        // 0,4,...,252

  v8f acc0 = {0.f,0.f,0.f,0.f,0.f,0.f,0.f,0.f};  // D tile, b = 0..15
  v8f acc1 = {0.f,0.f,0.f,0.f,0.f,0.f,0.f,0.f};  // D tile, b = 16..31

  for (int ib = 0; ib < IN_ / IBLK; ++ib) {
    const int i0 = ib * IBLK;
    __syncthreads();  // previous iteration's LDS consumers done

    // Stage aggregated weights: 16 rows x 256 cols, float4 per lane.
    // Per (k, pass): 64 lanes x 16B = contiguous 1KB -> global_load_b128.
    #pragma unroll
    for (int p = 0; p < OTILE / 4; ++p) {
      const int r = p * 4 + srow;
      const size_t base = (((size_t)(oBase + r)) * IN_ + i0 + scol) / 4;
      float4 s = {0.f, 0.f, 0.f, 0.f};
      #pragma unroll
      for (int k = 0; k < K_; ++k) {
        const float4 wv4 = ((const float4*)kw)[(size_t)k * (OUT_ * IN_ / 4) + base];
        const float ak = a_sm[k];
        s.x = fmaf(ak, wv4.x, s.x);
        s.y = fmaf(ak, wv4.y, s.y);
        s.z = fmaf(ak, wv4.z, s.z);
        s.w = fmaf(ak, wv4.w, s.w);
      }
      *(float4*)&aW[r][scol] = s;         // ds_store_b128 (16B aligned)
    }
    // Stage x slice: 32 rows x 256 cols, float4 per lane.
    #pragma unroll
    for (int p = 0; p < B_ / 4; ++p) {
      const int r = p * 4 + srow;
      *(float4*)&xs[r][scol] =
          ((const float4*)x)[((size_t)r * IN_ + i0 + scol) / 4];
    }
    __syncthreads();

    // WMMA over this wave's 32-column K slice.
    // 32-bit A layout (ISA 7.12.2): lane%16 = M row; VGPR v holds K = 2*(lane/16)+v.
    // B mirrors it with lane%16 = N column.
    const int cbase = ks * 32;
    #pragma unroll
    for (int ic = 0; ic < 32; ic += 4) {
      const int col = cbase + ic + 2 * sel;
      v2f av  = *(const v2f*)&aW[row][col];
      v2f bv0 = *(const v2f*)&xs[row][col];
      v2f bv1 = *(const v2f*)&xs[16 + row][col];
      acc0 = __builtin_amdgcn_wmma_f32_16x16x4_f32(false, av, false, bv0,
                                                   (short)0, acc0, false, false);
      acc1 = __builtin_amdgcn_wmma_f32_16x16x4_f32(false, av, false, bv1,
                                                   (short)0, acc1, false, false);
    }
  }

  // Deterministic cross-wave K reduction through LDS.
  __syncthreads();
  #pragma unroll
  for (int r = 0; r < 8; ++r) {
    red[wv * 512 +       r * 32 + ln] = acc0[r];
    red[wv * 512 + 256 + r * 32 + ln] = acc1[r];
  }
  __syncthreads();

  // 512 outputs (16 o x 32 b): thread t owns batch row bb = t>>3 and two
  // consecutive o's -> coalesced global_store_b64.
  {
    const int bb   = tid >> 3;            // 0..31
    const int ocol = (tid & 7) * 2;       // 0,2,...,14
    const int nt   = bb >> 4;
    const int lb   = bb & 15;
    float2 res;
    float* pr = &res.x;
    #pragma unroll
    for (int j = 0; j < 2; ++j) {
      const int ol = ocol + j;            // o_local 0..15
      const int r  = ol & 7;
      const int hi = ol >> 3;
      const int l  = hi * 16 + lb;        // lane of D element
      float s = 0.f;
      #pragma unroll
      for (int w = 0; w < 8; ++w)         // fixed order -> deterministic
        s += red[w * 512 + nt * 256 + r * 32 + l];
      pr[j] = s + ws[WS_AGGB + oBase + ol];
    }
    *(float2*)&out[(size_t)bb * OUT_ + oBase + ocol] = res;
  }
}

// ---------------------------------------------------------------------------
extern "C" void kernel_launch(void* const* d_in, const int* in_sizes, int n_in,
                              void* d_out, int out_size, void* d_ws, size_t ws_size,
                              hipStream_t stream) {
  (void)in_sizes; (void)n_in; (void)out_size; (void)ws_size;
  const float* x    = (const float*)d_in[0];  // (32, 2048)
  const float* cond = (const float*)d_in[1];  // (1, 2048)
  const float* w1   = (const float*)d_in[2];  // (2048, 512)
  const float* b1   = (const float*)d_in[3];  // (512)
  const float* w2   = (const float*)d_in[4];  // (512, 8)
  const float* b2   = (const float*)d_in[5];  // (8)
  const float* kw   = (const float*)d_in[6];  // (8, 2048, 2048)
  const float* kb   = (const float*)d_in[7];  // (8, 2048)
  float* out = (float*)d_out;                 // (32, 2048)
  float* ws  = (float*)d_ws;                  // needs 10248 floats (~41 KB)

  stage1_hpart<<<16, 512, 0, stream>>>(cond, w1, ws);
  stage2_scores<<<1, 512, 0, stream>>>(b1, w2, b2, kb, ws);
  moe_gemm<<<OUT_ / OTILE, 256, 0, stream>>>(x, kw, ws, out);
}